// GraphAttentionNetwork_87033217286589
// MI455X (gfx1250) — compile-verified
//
#include <hip/hip_runtime.h>
#include <math.h>

typedef __attribute__((ext_vector_type(2))) float v2f;
typedef __attribute__((ext_vector_type(8))) float v8f;

#define HEADS 4
#define NEG_SLOPE 0.2f

__device__ __forceinline__ unsigned flipf(float f) {
  unsigned u = __float_as_uint(f);
  return u ^ ((unsigned)((int)u >> 31) | 0x80000000u);
}
__device__ __forceinline__ float unflipf(unsigned u) {
  return __uint_as_float(u ^ ((unsigned)((int)(~u) >> 31) | 0x80000000u));
}
__device__ __forceinline__ float lrelu(float v) { return v > 0.f ? v : NEG_SLOPE * v; }

// ---------------------------------------------------------------------------
// Y[M,Nf] = X[M,K] @ W[K,Nf], fp32, one 16x16 tile per wave via
// V_WMMA_F32_16X16X4_F32 (K stepped by 4, 32 WMMAs for K=128).
// ISA 7.12.2 layouts: A 16x4 -> lanes 0-15 hold K=k..k+1 (VGPR0/1),
// lanes 16-31 hold K=k+2..k+3.  B 4x16 mirrored over N.  C/D: VGPR v holds
// row M = v + 8*(lane>=16), col N = lane&15.
// Requires M%16==0, Nf%64==0, K%4==0 (true here: M=50000, Nf=128/256, K=128).
// ---------------------------------------------------------------------------
__global__ void gemm_wmma_f32(const float* __restrict__ X, const float* __restrict__ Wm,
                              float* __restrict__ Y, int K, int Nf) {
  const int lane  = threadIdx.x & 31;
  const int wave  = threadIdx.x >> 5;
  const int tileM = blockIdx.x << 4;
  const int tileN = ((blockIdx.y << 2) + wave) << 4;
  const int l15   = lane & 15;
  const int khalf = (lane >> 4) << 1;  // 0 or 2

  const float* __restrict__ xrow = X + (size_t)(tileM + l15) * K;
  const float* __restrict__ wcol = Wm + (tileN + l15);

  v8f acc = {};
#pragma unroll 4
  for (int k = 0; k < K; k += 4) {
    v2f a = *(const v2f*)(xrow + k + khalf);  // A frag: (M=l15, K=k+khalf..+1)
    v2f b;
    b.x = wcol[(size_t)(k + khalf) * Nf];      // B frag: (K=k+khalf,   N=l15)
    b.y = wcol[(size_t)(k + khalf + 1) * Nf];  //         (K=k+khalf+1, N=l15)
    acc = __builtin_amdgcn_wmma_f32_16x16x4_f32(false, a, false, b, (short)0, acc,
                                                false, false);
  }
  const int colOut  = tileN + l15;
  const int rowBase = tileM + ((lane >> 4) << 3);
  float* __restrict__ yp = Y + (size_t)rowBase * Nf + colOut;
#pragma unroll
  for (int v = 0; v < 8; ++v) yp[(size_t)v * Nf] = acc[v];
}

// alpha_src[n,h] = sum_c h[n,h,c]*a_src[h,c]  (and same for a_dst)
__global__ void alpha_kernel(const float* __restrict__ h, const float* __restrict__ a_s,
                             const float* __restrict__ a_d, float* __restrict__ asrc,
                             float* __restrict__ adst, int N, int Cc) {
  int t = blockIdx.x * blockDim.x + threadIdx.x;
  if (t >= N * HEADS) return;
  int n = t >> 2, hd = t & 3;
  const float* hp = h + ((size_t)n * HEADS + hd) * Cc;
  const float* as = a_s + hd * Cc;
  const float* ad = a_d + hd * Cc;
  float s1 = 0.f, s2 = 0.f;
  for (int c = 0; c < Cc; ++c) { float v = hp[c]; s1 += v * as[c]; s2 += v * ad[c]; }
  asrc[t] = s1; adst[t] = s2;
}

__global__ void init_stats_kernel(unsigned* __restrict__ emaxu, float* __restrict__ denom,
                                  int n) {
  int t = blockIdx.x * blockDim.x + threadIdx.x;
  if (t < n) { emaxu[t] = 0u; denom[t] = 0.f; }  // flipped-order 0 == -inf
}

// out[n, :] = bias (or 0)
__global__ void init_out_kernel(float* __restrict__ out, const float* __restrict__ b) {
  size_t i = (size_t)blockIdx.x * blockDim.x + threadIdx.x;
  out[i] = b ? b[threadIdx.x] : 0.f;
}

// Edges e<E come from edge_index; e>=E are the self-loops (e-E, e-E).
__global__ void edge_max_kernel(const int* __restrict__ src, const int* __restrict__ dst,
                                int E, int N, const float* __restrict__ asrc,
                                const float* __restrict__ adst, unsigned* __restrict__ emaxu) {
  long long t = (long long)blockIdx.x * blockDim.x + threadIdx.x;
  if (t >= (long long)(E + N) * HEADS) return;
  int e = (int)(t >> 2), hd = (int)(t & 3);
  int s = (e < E) ? src[e] : (e - E);
  int d = (e < E) ? dst[e] : (e - E);
  float v = lrelu(asrc[s * HEADS + hd] + adst[d * HEADS + hd]);
  atomicMax(&emaxu[d * HEADS + hd], flipf(v));  // global_atomic_max_u32
}

__global__ void edge_sum_kernel(const int* __restrict__ src, const int* __restrict__ dst,
                                int E, int N, const float* __restrict__ asrc,
                                const float* __restrict__ adst,
                                const unsigned* __restrict__ emaxu, float* __restrict__ denom) {
  long long t = (long long)blockIdx.x * blockDim.x + threadIdx.x;
  if (t >= (long long)(E + N) * HEADS) return;
  int e = (int)(t >> 2), hd = (int)(t & 3);
  int s = (e < E) ? src[e] : (e - E);
  int d = (e < E) ? dst[e] : (e - E);
  int idx = d * HEADS + hd;
  float v = lrelu(asrc[s * HEADS + hd] + adst[idx]);
  atomicAdd(&denom[idx], expf(v - unflipf(emaxu[idx])));
}

// One block per edge, blockDim = H*C channels -> coalesced gather + scatter.
__global__ void aggregate_kernel(const int* __restrict__ src, const int* __restrict__ dst,
                                 int E, int cshift, const float* __restrict__ h,
                                 const float* __restrict__ asrc, const float* __restrict__ adst,
                                 const unsigned* __restrict__ emaxu,
                                 const float* __restrict__ denom, float* __restrict__ out) {
  int e = blockIdx.x;
  int j = threadIdx.x;
  int F = blockDim.x;
  int hd = j >> cshift;
  int s = (e < E) ? src[e] : (e - E);
  int d = (e < E) ? dst[e] : (e - E);
  int idx = d * HEADS + hd;
  float v = lrelu(asrc[s * HEADS + hd] + adst[idx]);
  float alpha = expf(v - unflipf(emaxu[idx])) / denom[idx];
  atomicAdd(&out[(size_t)d * F + j], h[(size_t)s * F + j] * alpha);
}

// layer-3: mean over 4 heads of [N,4,64] + bias -> [N,64]
__global__ void finalize_mean_kernel(const float* __restrict__ acc, const float* __restrict__ b,
                                     float* __restrict__ out, int N) {
  int t = blockIdx.x * blockDim.x + threadIdx.x;
  if (t >= N * 64) return;
  int n = t >> 6, c = t & 63;
  const float* ap = acc + (size_t)n * 256;
  out[t] = (ap[c] + ap[64 + c] + ap[128 + c] + ap[192 + c]) * 0.25f + b[c];
}

// ---------------------------------------------------------------------------
static void run_gat_layer(hipStream_t stream, const float* xin, int K, const float* Wm,
                          const float* a_s, const float* a_d, const float* bias, int Cc,
                          int cshift, int N, const int* src, const int* dst, int E,
                          float* h, float* out, float* asrc, float* adst,
                          unsigned* emaxu, float* denom) {
  const int F = HEADS * Cc;
  dim3 gg(N / 16, F / 64);
  gemm_wmma_f32<<<gg, 128, 0, stream>>>(xin, Wm, h, K, F);

  int NH = N * HEADS;
  alpha_kernel<<<(NH + 255) / 256, 256, 0, stream>>>(h, a_s, a_d, asrc, adst, N, Cc);
  init_stats_kernel<<<(NH + 255) / 256, 256, 0, stream>>>(emaxu, denom, NH);

  long long EH = (long long)(E + N) * HEADS;
  int eb = (int)((EH + 255) / 256);
  edge_max_kernel<<<eb, 256, 0, stream>>>(src, dst, E, N, asrc, adst, emaxu);
  edge_sum_kernel<<<eb, 256, 0, stream>>>(src, dst, E, N, asrc, adst, emaxu, denom);

  init_out_kernel<<<N, F, 0, stream>>>(out, bias);
  aggregate_kernel<<<E + N, F, 0, stream>>>(src, dst, E, cshift, h, asrc, adst,
                                            emaxu, denom, out);
}

extern "C" void kernel_launch(void* const* d_in, const int* in_sizes, int n_in,
                              void* d_out, int out_size, void* d_ws, size_t ws_size,
                              hipStream_t stream) {
  const float* x   = (const float*)d_in[0];
  const int*   ei  = (const int*)d_in[1];
  const float* W1  = (const float*)d_in[2];
  const float* as1 = (const float*)d_in[3];
  const float* ad1 = (const float*)d_in[4];
  const float* b1  = (const float*)d_in[5];
  const float* W2  = (const float*)d_in[6];
  const float* as2 = (const float*)d_in[7];
  const float* ad2 = (const float*)d_in[8];
  const float* b2  = (const float*)d_in[9];
  const float* W3  = (const float*)d_in[10];
  const float* as3 = (const float*)d_in[11];
  const float* ad3 = (const float*)d_in[12];
  const float* b3  = (const float*)d_in[13];

  const int N = in_sizes[0] / 128;  // 50000
  const int E = in_sizes[1] / 2;    // 800000
  const int* src = ei;
  const int* dst = ei + E;

  // workspace layout (~106 MB)
  float*    bufH  = (float*)d_ws;                      // GEMM output  [N,256] max
  float*    bufO  = bufH + (size_t)N * 256;            // layer output [N,256] max
  float*    asrc  = bufO + (size_t)N * 256;            // [N,4]
  float*    adst  = asrc + (size_t)N * HEADS;          // [N,4]
  unsigned* emaxu = (unsigned*)(adst + (size_t)N * HEADS);  // [N,4]
  float*    denom = (float*)(emaxu + (size_t)N * HEADS);    // [N,4]

  // Layer 1: x[N,128] -> bufO[N,128] (concat, +b1)
  run_gat_layer(stream, x, 128, W1, as1, ad1, b1, 32, 5, N, src, dst, E,
                bufH, bufO, asrc, adst, emaxu, denom);
  // Layer 2: bufO[N,128] -> bufO[N,128] (GEMM into bufH first, so safe)
  run_gat_layer(stream, bufO, 128, W2, as2, ad2, b2, 32, 5, N, src, dst, E,
                bufH, bufO, asrc, adst, emaxu, denom);
  // Layer 3: bufO[N,128] -> per-head acc in bufO[N,256] (no bias yet)
  run_gat_layer(stream, bufO, 128, W3, as3, ad3, nullptr, 64, 6, N, src, dst, E,
                bufH, bufO, asrc, adst, emaxu, denom);
  // Mean over heads + b3 -> d_out [N,64]
  finalize_mean_kernel<<<(N * 64 + 255) / 256, 256, 0, stream>>>(bufO, b3, (float*)d_out, N);
}